// DownwardPropagationDirect_86560770883612
// MI455X (gfx1250) — compile-verified
//
#include <hip/hip_runtime.h>
#include <cstdint>

#define Bn 131072
#define Ln 60
#define Cn 48

#define TB_BYTES 11520            // 60*48*4 bytes of t_direct per batch
#define CHUNK_L 12                // layers staged per chunk
#define CHUNK_B1 2304             // 12*48*4 bytes per batch per chunk
#define CHUNK_BYTES 4608          // two batches per wave
#define NCHUNK 5                  // 60 / 12
#define WAVES_PER_BLOCK 8
#define LDS_FLOATS (WAVES_PER_BLOCK * 2 * (CHUNK_BYTES / 4))   // 18432 floats = 72 KB

// DPP controls
#define DPP_QUAD_XOR1  0xB1   // quad_perm(1,0,3,2)
#define DPP_QUAD_XOR2  0x4E   // quad_perm(2,3,0,1)
#define DPP_ROW_HMIRR  0x141  // row_half_mirror (lane ^ 7 within 8)
#define DPP_ROW_MIRROR 0x140  // row_mirror      (lane ^ 15 within 16)

template <int CTRL>
__device__ __forceinline__ float dpp_add(float v) {
  int x = __builtin_amdgcn_update_dpp(0, __float_as_int(v), CTRL, 0xf, 0xf, true);
  return v + __int_as_float(x);
}

// Sum across each 16-lane row of the wave32; result broadcast to all 16 lanes.
// DPP rows are 16 lanes, so the two batch halves never mix.
__device__ __forceinline__ float reduce16(float v) {
  v = dpp_add<DPP_QUAD_XOR1>(v);
  v = dpp_add<DPP_QUAD_XOR2>(v);
  v = dpp_add<DPP_ROW_HMIRR>(v);
  v = dpp_add<DPP_ROW_MIRROR>(v);
  return v;
}

// Stage one chunk (12 layers x 2 batches = 4608 B) into LDS with nine
// fully-coalesced async b128 loads (32 lanes x 16 B each). Flat offsets
// [0,2304) belong to batch b0, [2304,4608) to batch b0+1 (regions TB_BYTES
// apart in global memory).
__device__ __forceinline__ void issue_chunk(uint64_t tb, uint32_t lds_base,
                                            int c, int lane) {
#pragma unroll
  for (int j = 0; j < 9; ++j) {
    uint32_t flat = (uint32_t)(j * 512) + (uint32_t)lane * 16u;
    uint32_t goff = (uint32_t)(c * CHUNK_B1) + flat +
                    ((flat >= (uint32_t)CHUNK_B1) ? (uint32_t)(TB_BYTES - CHUNK_B1) : 0u);
    uint64_t ga = tb + (uint64_t)goff;
    uint32_t la = lds_base + flat;
    asm volatile("global_load_async_to_lds_b128 %0, %1, off"
                 :: "v"(la), "v"(ga)
                 : "memory");
  }
}

__global__ __launch_bounds__(256) void ddp_kernel(const float* __restrict__ F,
                                                  const float* __restrict__ T,
                                                  float* __restrict__ out) {
  __shared__ __align__(16) float smem[LDS_FLOATS];

  const int tid  = (int)threadIdx.x;
  const int wave = tid >> 5;
  const int lane = tid & 31;
  const int half = lane >> 4;   // which of the wave's two batches
  const int li   = lane & 15;   // channel group: channels 3*li .. 3*li+2

  const int b0 = ((int)blockIdx.x * WAVES_PER_BLOCK + wave) * 2;
  const int b  = b0 + half;

  const float* fp = F + (size_t)b * Cn + li * 3;
  const float f0 = fp[0], f1 = fp[1], f2 = fp[2];

  float* ob = out + (size_t)b * (Ln + 1);

  // Layer 0: sum of input_flux over channels (broadcast to all 16 lanes).
  const float s0 = reduce16(f0 + f1 + f2);

  // LDS byte offset of smem (flat address low 32 bits == LDS offset).
  const uint32_t smem_off = (uint32_t)(uintptr_t)(&smem[0]);
  const uint32_t wb = smem_off + (uint32_t)wave * (2u * CHUNK_BYTES);
  const uint64_t tb = (uint64_t)(uintptr_t)T + (uint64_t)b0 * TB_BYTES;

  // Prime the double buffer.
  issue_chunk(tb, wb, 0, lane);
  issue_chunk(tb, wb + CHUNK_BYTES, 1, lane);

  float r0 = 1.0f, r1 = 1.0f, r2 = 1.0f;

  // Each lane keeps the reduced sum for layers lg with lg % 16 == li.
  float keep0 = 0.0f, keep1 = 0.0f, keep2 = 0.0f, keep3 = 0.0f;

  const int rbase0 = wave * 2304 + half * 576 + li * 3;   // float index into smem

#pragma unroll
  for (int c = 0; c < NCHUNK; ++c) {
    // Chunk c landed (9 loads of chunk c+1 may still be in flight).
    if (c < NCHUNK - 1)
      asm volatile("s_wait_asynccnt 9" ::: "memory");
    else
      asm volatile("s_wait_asynccnt 0" ::: "memory");

    const int rb = rbase0 + (c & 1) * 1152;
#pragma unroll
    for (int l = 0; l < CHUNK_L; ++l) {
      const int lg = c * CHUNK_L + l;                 // global layer, compile-time
      const float t0 = smem[rb + l * 48 + 0];
      const float t1 = smem[rb + l * 48 + 1];
      const float t2 = smem[rb + l * 48 + 2];
      r0 *= t0;
      r1 *= t1;
      r2 *= t2;
      float p = fmaf(f0, r0, fmaf(f1, r1, f2 * r2));
      p = reduce16(p);                                // broadcast to all 16 lanes
      const bool mine = (li == (lg & 15));            // one v_cndmask per layer
      if ((lg >> 4) == 0)      keep0 = mine ? p : keep0;
      else if ((lg >> 4) == 1) keep1 = mine ? p : keep1;
      else if ((lg >> 4) == 2) keep2 = mine ? p : keep2;
      else                     keep3 = mine ? p : keep3;
    }

    if (c + 2 < NCHUNK) {
      // All LDS reads of this buffer retired before the async DMA overwrites it.
      asm volatile("s_wait_dscnt 0" ::: "memory");
      issue_chunk(tb, wb + (c & 1) * CHUNK_BYTES, c + 2, lane);
    }
  }

  // Coalesced epilogue stores: 16 consecutive floats per row, both halves in
  // one instruction. Covers ob[0..60] exactly once.
  ob[1 + li]  = keep0;   // layers 0..15
  ob[17 + li] = keep1;   // layers 16..31
  ob[33 + li] = keep2;   // layers 32..47
  if (li < 12) ob[49 + li] = keep3;   // layers 48..59
  if (li == 0) ob[0] = s0;            // input flux sum
}

extern "C" void kernel_launch(void* const* d_in, const int* in_sizes, int n_in,
                              void* d_out, int out_size, void* d_ws, size_t ws_size,
                              hipStream_t stream) {
  const float* F = (const float*)d_in[0];   // input_flux [B,48] f32
  const float* T = (const float*)d_in[1];   // t_direct  [B,60,48] f32
  float* out = (float*)d_out;               // [B,61] f32

  dim3 grid(Bn / (2 * WAVES_PER_BLOCK));    // 8192 blocks
  dim3 block(256);                          // 8 waves, 2 batches per wave
  hipLaunchKernelGGL(ddp_kernel, grid, block, 0, stream, F, T, out);
}